// Detect_15126874817007
// MI455X (gfx1250) — compile-verified
//
#include <hip/hip_runtime.h>
#include <math.h>

typedef float v2f __attribute__((ext_vector_type(2)));
typedef float v8f __attribute__((ext_vector_type(8)));

constexpr int   BB          = 8;
constexpr int   NN          = 32768;
constexpr int   NCLS        = 81;
constexpr int   NC          = 80;      // foreground classes
constexpr int   MTOP        = 200;     // TOP_K
constexpr int   KKEEP       = 200;     // KEEP_TOP_K
constexpr float CONF_TH     = 0.01f;
constexpr float NMS_TH      = 0.45f;
constexpr int   CAND        = 1024;

#if defined(__gfx1250__)
#define HAVE_ASYNC_LDS 1
#else
#define HAVE_ASYNC_LDS 0
#endif

#if HAVE_ASYNC_LDS
// 16B global -> LDS async copy (VFLAT async op, tracked by ASYNCcnt).
// First operand is the LDS byte address; for aperture-based flat shared
// pointers the LDS offset is addr[31:0] (ISA 10.2 aperture mapping).
__device__ __forceinline__ void async_cp16(const float* g, const float* l) {
  unsigned           lds_off = (unsigned)(size_t)l;
  unsigned long long gaddr   = (unsigned long long)(size_t)g;
  asm volatile("global_load_async_to_lds_b128 %0, %1, off"
               :: "v"(lds_off), "v"(gaddr)
               : "memory");
}
__device__ __forceinline__ void wait_async() {
  asm volatile("s_wait_asynccnt 0" ::: "memory");
}
#endif

// ---------------------------------------------------------------------------
// Kernel 1: decode boxes + per-anchor softmax stats (max, sum-exp).
// 128 threads = 4 waves; each block owns 64 anchors of one batch.
// Conf tile (64 x 81 f32 = 20.7KB) is streamed into LDS with async copies,
// overlapped with the box decode which reads independent data.
// Sum-of-exp runs on the XDL pipe: V_WMMA_F32_16X16X4_F32 with an all-ones
// B matrix reduces 16 anchors x 4 classes per instruction, co-executing with
// the VALU exp stream.
// ---------------------------------------------------------------------------
__global__ __launch_bounds__(128) void k_decode_stats(
    const float* __restrict__ loc, const float* __restrict__ conf,
    const float* __restrict__ dbox, float* __restrict__ boxes,
    float* __restrict__ stats) {
  __shared__ float s_conf[64 * 81];

  const int blocksPerB = NN / 64;  // 512
  const int b   = blockIdx.x / blocksPerB;
  const int n0  = (blockIdx.x % blocksPerB) * 64;
  const int tid = threadIdx.x;

  const float* gsrc = conf + ((size_t)b * NN + n0) * NCLS;
  // 64*81 floats = 1296 chunks of 16B
#if HAVE_ASYNC_LDS
  for (int idx = tid; idx < 1296; idx += 128)
    async_cp16(gsrc + idx * 4, s_conf + idx * 4);
#else
  for (int idx = tid; idx < 1296; idx += 128)
    ((float4*)s_conf)[idx] = ((const float4*)gsrc)[idx];
#endif

  // Box decode overlaps with the async conf copy (independent data).
  if (tid < 64) {
    const int n = n0 + tid;
    float4 l = ((const float4*)loc)[(size_t)b * NN + n];
    float4 d = ((const float4*)dbox)[n];
    float cx = d.x + l.x * 0.1f * d.z;
    float cy = d.y + l.y * 0.1f * d.w;
    float w  = d.z * __expf(l.z * 0.2f);
    float h  = d.w * __expf(l.w * 0.2f);
    float x1 = cx - 0.5f * w, y1 = cy - 0.5f * h;
    float x2 = x1 + w,        y2 = y1 + h;
    float4 bx;
    bx.x = fminf(fmaxf(x1, 0.f), 1.f);
    bx.y = fminf(fmaxf(y1, 0.f), 1.f);
    bx.z = fminf(fmaxf(x2, 0.f), 1.f);
    bx.w = fminf(fmaxf(y2, 0.f), 1.f);
    ((float4*)boxes)[(size_t)b * NN + n] = bx;
  }

#if HAVE_ASYNC_LDS
  wait_async();
#endif
  __syncthreads();

  const int lane = tid & 31;
  const int wv   = tid >> 5;
  const int al   = wv * 16 + (lane & 15);  // local anchor 0..63 (16 per wave)

  // Row max: lanes split even/odd classes, combine across the half-waves.
  float m = -INFINITY;
  for (int c = (lane >> 4); c < NCLS; c += 2)
    m = fmaxf(m, s_conf[al * NCLS + c]);
  m = fmaxf(m, __shfl_xor(m, 16, 32));

  // Sum of exp via WMMA: A = 16x4 tile of exp values, B = ones(4x16).
  v8f acc = {0.f, 0.f, 0.f, 0.f, 0.f, 0.f, 0.f, 0.f};
  v2f bones = {1.f, 1.f};
  const int koff = (lane >> 4) << 1;  // K = 0/1 (lanes 0-15), 2/3 (lanes 16-31)
  for (int kb = 0; kb < 84; kb += 4) {
    const int c0 = kb + koff;
    v2f a;
    a.x = (c0     < NCLS) ? __expf(s_conf[al * NCLS + c0    ] - m) : 0.f;
    a.y = (c0 + 1 < NCLS) ? __expf(s_conf[al * NCLS + c0 + 1] - m) : 0.f;
    acc = __builtin_amdgcn_wmma_f32_16x16x4_f32(false, a, false, bones,
                                                (short)0, acc, false, false);
  }
  // D layout: rows 0-7 in c[0..7] of lanes 0-15, rows 8-15 in lanes 16-31;
  // B==ones makes all columns identical.
  const int ri = lane & 7;
  float v  = acc[ri];
  float w2 = __shfl(v, ri + 16, 32);
  float ssum = (lane < 8) ? v : w2;
  if (lane < 16) {
    const int n = n0 + al;
    ((float2*)stats)[(size_t)b * NN + n] = make_float2(m, ssum);
  }
}

// ---------------------------------------------------------------------------
// Kernel 2: one block per (batch, class): top-200 selection + greedy NMS.
// Scores quantized to u16 keys cached in LDS; 16-step integer binary search
// finds the 200th-largest key; bounded collect; bitonic sort (desc, idx-asc
// tie break, matching lax.top_k); NMS matches !(iou <= thresh) semantics.
// ---------------------------------------------------------------------------
__global__ __launch_bounds__(256) void k_topk_nms(
    const float* __restrict__ conf, const float* __restrict__ stats,
    const float* __restrict__ boxes, float* __restrict__ kept,
    float* __restrict__ cbox) {
  __shared__ unsigned short s_key[NN];   // 64KB of 320KB/WGP LDS
  __shared__ int    s_red[256];
  __shared__ float  s_cs[CAND];
  __shared__ int    s_ci[CAND];
  __shared__ float4 s_bx[MTOP];
  __shared__ float  s_sc2[MTOP];
  __shared__ int    s_act[MTOP];
  __shared__ int    s_keep[MTOP];
  __shared__ int    s_cnt;

  const int b   = blockIdx.x / NC;
  const int cls = blockIdx.x % NC;
  const int tid = threadIdx.x;

  // Pass 1: quantized score keys (0 = below threshold).
  for (int n = tid; n < NN; n += 256) {
    float2 st = ((const float2*)stats)[(size_t)b * NN + n];
    float cv  = conf[((size_t)b * NN + n) * NCLS + (cls + 1)];
    float sc  = __expf(cv - st.x) / st.y;
    int key = 0;
    if (sc > CONF_TH) {
      int q = (int)(sc * 65535.0f);
      key = (q > 65534 ? 65534 : q) + 1;
    }
    s_key[n] = (unsigned short)key;
  }
  __syncthreads();

  // Binary search over integer key space for the top-200 threshold.
  int lo = 1, hi = 65536;
  for (int it = 0; it < 16; ++it) {
    const int mid = (lo + hi) >> 1;
    int partial = 0;
    for (int n = tid; n < NN; n += 256) partial += (s_key[n] >= mid);
    s_red[tid] = partial;
    __syncthreads();
    for (int off = 128; off > 0; off >>= 1) {
      if (tid < off) s_red[tid] += s_red[tid + off];
      __syncthreads();
    }
    const int cnt = s_red[0];
    __syncthreads();
    if (cnt >= MTOP) lo = mid; else hi = mid;
  }
  const int kthr = lo;

  // Collect candidates (exact float score recomputed for hits only).
  if (tid == 0) s_cnt = 0;
  __syncthreads();
  for (int n = tid; n < NN; n += 256) {
    if ((int)s_key[n] >= kthr) {
      const int pos = atomicAdd(&s_cnt, 1);
      if (pos < CAND) {
        float2 st = ((const float2*)stats)[(size_t)b * NN + n];
        float cv  = conf[((size_t)b * NN + n) * NCLS + (cls + 1)];
        s_cs[pos] = __expf(cv - st.x) / st.y;
        s_ci[pos] = n;
      }
    }
  }
  __syncthreads();
  int cnt0 = s_cnt;
  if (cnt0 > CAND) cnt0 = CAND;
  for (int i = tid; i < CAND; i += 256)
    if (i >= cnt0) { s_cs[i] = -1.0f; s_ci[i] = 0x7FFFFFFF; }
  __syncthreads();

  // Bitonic sort: descending score, ascending index on ties (deterministic).
  for (int kk = 2; kk <= CAND; kk <<= 1) {
    for (int j = kk >> 1; j > 0; j >>= 1) {
      for (int i = tid; i < CAND; i += 256) {
        const int ixj = i ^ j;
        if (ixj > i) {
          float s0 = s_cs[i], s1 = s_cs[ixj];
          int   i0 = s_ci[i], i1 = s_ci[ixj];
          const bool desc = ((i & kk) == 0);
          const bool sw = desc ? (s1 > s0 || (s1 == s0 && i1 < i0))
                               : (s0 > s1 || (s0 == s1 && i0 < i1));
          if (sw) { s_cs[i] = s1; s_cs[ixj] = s0; s_ci[i] = i1; s_ci[ixj] = i0; }
        }
      }
      __syncthreads();
    }
  }

  // Gather boxes of the top-200 and run greedy NMS.
  if (tid < MTOP) {
    const float sc = s_cs[tid];
    int n = s_ci[tid];
    if ((unsigned)n >= (unsigned)NN) n = 0;  // -1 pads: any valid box, inactive
    s_bx[tid]  = ((const float4*)boxes)[(size_t)b * NN + n];
    s_sc2[tid] = sc;
    s_act[tid] = (sc > CONF_TH) ? 1 : 0;
  }
  __syncthreads();

  for (int i = 0; i < MTOP; ++i) {
    const int    ki = s_act[i];
    const float4 bi = s_bx[i];
    __syncthreads();  // everyone reads s_act[i] before any suppression write
    if (ki && tid < MTOP && s_act[tid]) {
      const float4 bj = s_bx[tid];
      float areaI = (bi.z - bi.x) * (bi.w - bi.y);
      float areaJ = (bj.z - bj.x) * (bj.w - bj.y);
      float xx1 = fmaxf(bi.x, bj.x), yy1 = fmaxf(bi.y, bj.y);
      float xx2 = fminf(bi.z, bj.z), yy2 = fminf(bi.w, bj.w);
      float inter = fmaxf(xx2 - xx1, 0.f) * fmaxf(yy2 - yy1, 0.f);
      float iou = inter / (areaJ - inter + areaI);
      if (!(iou <= NMS_TH)) s_act[tid] = 0;  // NaN suppresses, like reference
    }
    if (tid == 0) s_keep[i] = ki;
    __syncthreads();
  }

  if (tid < MTOP) {
    const size_t base = ((size_t)b * NC + cls) * MTOP + tid;
    kept[base] = s_keep[tid] ? s_sc2[tid] : 0.0f;
    ((float4*)cbox)[base] = s_bx[tid];
  }
}

// ---------------------------------------------------------------------------
// Kernel 3: per batch — global rank (stable, matches argsort semantics),
// keep rank < 200, per-class stable compaction into [81,200,5] output.
// ---------------------------------------------------------------------------
__global__ __launch_bounds__(256) void k_global(
    const float* __restrict__ kept, const float* __restrict__ cbox,
    float* __restrict__ out) {
  constexpr int F = NC * MTOP;  // 16000
  __shared__ float    s_flat[F];
  __shared__ unsigned s_kb[(F + 31) / 32];

  const int b   = blockIdx.x;
  const int tid = threadIdx.x;

  for (int p = tid; p < F; p += 256) s_flat[p] = kept[(size_t)b * F + p];
  for (int i = tid; i < (F + 31) / 32; i += 256) s_kb[i] = 0u;
  for (int i = tid; i < NCLS * KKEEP * 5; i += 256)
    out[(size_t)b * NCLS * KKEEP * 5 + i] = 0.0f;
  __syncthreads();

  for (int p = tid; p < F; p += 256) {
    const float s = s_flat[p];
    if (s > 0.0f) {
      int rank = 0;
      for (int q = 0; q < F; ++q) {
        const float v = s_flat[q];
        rank += (v > s) || (v == s && q < p);
      }
      if (rank < KKEEP) atomicOr(&s_kb[p >> 5], 1u << (p & 31));
    }
  }
  __syncthreads();

  if (tid < NC) {
    const int c = tid;
    int slot = 0;
    for (int j = 0; j < MTOP; ++j) {
      const int p = c * MTOP + j;
      if (s_kb[p >> 5] & (1u << (p & 31))) {
        const size_t src = ((size_t)b * NC + c) * MTOP + j;
        const float4 bx = ((const float4*)cbox)[src];
        const size_t dst = (((size_t)b * NCLS + (c + 1)) * KKEEP + slot) * 5;
        out[dst]     = s_flat[p];
        out[dst + 1] = bx.x;
        out[dst + 2] = bx.y;
        out[dst + 3] = bx.z;
        out[dst + 4] = bx.w;
        ++slot;
      }
    }
  }
}

extern "C" void kernel_launch(void* const* d_in, const int* in_sizes, int n_in,
                              void* d_out, int out_size, void* d_ws,
                              size_t ws_size, hipStream_t stream) {
  (void)in_sizes; (void)n_in; (void)out_size; (void)ws_size;
  const float* loc  = (const float*)d_in[0];
  const float* conf = (const float*)d_in[1];
  const float* dbox = (const float*)d_in[2];
  float* out = (float*)d_out;

  float* ws    = (float*)d_ws;                      // ~8.9 MB used
  float* boxes = ws;                                // B*N*4
  float* stats = boxes + (size_t)BB * NN * 4;       // B*N*2  (max, sumexp)
  float* kept  = stats + (size_t)BB * NN * 2;       // B*80*200
  float* cbox  = kept  + (size_t)BB * NC * MTOP;    // B*80*200*4

  k_decode_stats<<<BB * (NN / 64), 128, 0, stream>>>(loc, conf, dbox, boxes,
                                                     stats);
  k_topk_nms<<<BB * NC, 256, 0, stream>>>(conf, stats, boxes, kept, cbox);
  k_global<<<BB, 256, 0, stream>>>(kept, cbox, out);
}